// QuanvolutionHybrid_65481071410220
// MI455X (gfx1250) — compile-verified
//
#include <hip/hip_runtime.h>
#include <math.h>

typedef __attribute__((ext_vector_type(2))) float v2f;
typedef __attribute__((ext_vector_type(8))) float v8f;

#define THRESH_F 0.8f
#define EPS_F    1e-7f

// ---- CDNA5 async global->LDS copy (ASYNCcnt-tracked), portable inline asm ----
__device__ __forceinline__ void async_copy_b128(unsigned ldsDst, unsigned byteOff,
                                                const void* base) {
  asm volatile("global_load_async_to_lds_b128 %0, %1, %2 offset:0"
               :: "v"(ldsDst), "v"(byteOff), "s"(base)
               : "memory");
}
__device__ __forceinline__ void wait_async0() {
  asm volatile("s_wait_asynccnt 0" ::: "memory");
}
__device__ __forceinline__ unsigned lds_addr(const void* p) {
  // generic(flat) LDS pointer: low 32 bits are the workgroup-relative LDS offset
  return (unsigned)(uintptr_t)p;
}

// ---------------------------------------------------------------------------
// Kernel 1: async-stage image -> LDS; 2x2/s2 conv + bias -> patches[196][4];
// normalize; 196x196 cosine sims via V_WMMA_F32_16X16X4_F32 (13x13 tiles);
// threshold >= 0.8; agg = adj @ patches / deg -> agg[B][784] (workspace).
// One 128-thread (4-wave) workgroup per image.
// ---------------------------------------------------------------------------
__launch_bounds__(128)
__global__ void k_quanv_agg(const float* __restrict__ x,
                            const float* __restrict__ cw,
                            const float* __restrict__ cb,
                            float* __restrict__ agg) {
  __shared__ float sX[784];                // staged 28x28 image
  __shared__ float sPatch[208 * 4];        // zero-padded rows 196..207
  __shared__ float sPn[208 * 4];           // normalized patches (padded)
  __shared__ float sStripe[4][16 * 208];   // per-wave 16-row sims stripe

  const int b   = blockIdx.x;
  const int tid = threadIdx.x;

  // ---- async stage of the image (3136 B = 196 x b128) ----
  {
    const void* xbase = (const void*)(x + (size_t)b * 784);
    for (int idx = tid; idx < 196; idx += 128)
      async_copy_b128(lds_addr(&sX[idx * 4]), (unsigned)(idx * 16), xbase);
    wait_async0();
  }
  __syncthreads();

  // ---- conv 2x2 stride 2 + bias; pad to 208 rows with zeros ----
  for (int idx = tid; idx < 208 * 4; idx += 128) {
    float v = 0.f;
    if (idx < 784) {
      const int n = idx >> 2, f = idx & 3;
      const int i = n / 14, j = n % 14;
      const float* xr = &sX[(2 * i) * 28 + 2 * j];
      v = cb[f]
        + xr[0]  * cw[f * 4 + 0]
        + xr[1]  * cw[f * 4 + 1]
        + xr[28] * cw[f * 4 + 2]
        + xr[29] * cw[f * 4 + 3];
    }
    sPatch[idx] = v;
  }
  __syncthreads();

  // ---- per-patch L2 normalization ----
  for (int n = tid; n < 208; n += 128) {
    const float p0 = sPatch[n * 4 + 0], p1 = sPatch[n * 4 + 1];
    const float p2 = sPatch[n * 4 + 2], p3 = sPatch[n * 4 + 3];
    const float nr  = sqrtf(p0 * p0 + p1 * p1 + p2 * p2 + p3 * p3);
    const float inv = 1.f / fmaxf(nr, EPS_F);
    sPn[n * 4 + 0] = p0 * inv; sPn[n * 4 + 1] = p1 * inv;
    sPn[n * 4 + 2] = p2 * inv; sPn[n * 4 + 3] = p3 * inv;
  }
  __syncthreads();

  const int wave = tid >> 5;
  const int lane = tid & 31;
  const int mrow = lane & 15;            // M (and N) index within tile
  const int ksel = (lane >> 4) * 2;      // K pair selector per ISA layout
  float* stripe  = &sStripe[wave][0];

  // 13 row-stripes of 16, distributed across the 4 waves (uniform per wave).
  for (int ti = wave; ti < 13; ti += 4) {
    v2f a;
    a.x = sPn[(ti * 16 + mrow) * 4 + ksel];
    a.y = sPn[(ti * 16 + mrow) * 4 + ksel + 1];

    for (int tj = 0; tj < 13; ++tj) {
      v2f bf;                            // B = A^T of same matrix: same fetch
      bf.x = sPn[(tj * 16 + mrow) * 4 + ksel];
      bf.y = sPn[(tj * 16 + mrow) * 4 + ksel + 1];
      v8f c = {};
      c = __builtin_amdgcn_wmma_f32_16x16x4_f32(false, a, false, bf,
                                                (short)0, c, false, false);
      // D layout: VGPR r -> row ((lane>=16)?8:0)+r, col = lane&15
      const int colBase = tj * 16 + mrow;
      const int rowBase = (lane >> 4) * 8;
#pragma unroll
      for (int r = 0; r < 8; ++r)
        stripe[(rowBase + r) * 208 + colBase] = c[r];
    }

    // Threshold + aggregate the 16 rows of this stripe.
    // Lane halves split the 196 columns; combine with shfl_xor(16) (wave32).
    const int row = mrow;
    const int mlo = (lane >> 4) * 98;
    float deg = 0.f, a0 = 0.f, a1 = 0.f, a2 = 0.f, a3 = 0.f;
    for (int m = mlo; m < mlo + 98; ++m) {
      const float s = stripe[row * 208 + m];
      if (s >= THRESH_F) {
        deg += 1.f;
        a0 += sPatch[m * 4 + 0]; a1 += sPatch[m * 4 + 1];
        a2 += sPatch[m * 4 + 2]; a3 += sPatch[m * 4 + 3];
      }
    }
    deg += __shfl_xor(deg, 16);
    a0  += __shfl_xor(a0, 16);
    a1  += __shfl_xor(a1, 16);
    a2  += __shfl_xor(a2, 16);
    a3  += __shfl_xor(a3, 16);

    const int n = ti * 16 + row;
    if (lane < 16 && n < 196) {
      const float inv = 1.f / fmaxf(deg, EPS_F);
      float4 o = make_float4(a0 * inv, a1 * inv, a2 * inv, a3 * inv);
      *(float4*)&agg[(size_t)b * 784 + n * 4] = o;
    }
  }
}

// ---------------------------------------------------------------------------
// Kernel 2 (fused MLP): async-stage agg tile [16,784] -> LDS; 8 waves cover
// the 8 column tiles of w1; h = relu(agg@w1+b1) kept in LDS; then layer 2
// (128x10) + log_softmax, writing out[16,10]. One 256-thread block per
// 16-image tile. K=784 swept with 4 independent WMMA accumulators for ILP.
// ---------------------------------------------------------------------------
__launch_bounds__(256)
__global__ void k_mlp_fused(const float* __restrict__ agg,
                            const float* __restrict__ w1,
                            const float* __restrict__ b1,
                            const float* __restrict__ w2,
                            const float* __restrict__ b2,
                            float* __restrict__ out) {
  __shared__ float sA[16 * 784];   // 50176 B activation tile
  __shared__ float sH[16 * 128];   // hidden tile
  __shared__ float sO[16 * 12];    // logits (padded stride 12)

  const int bt  = blockIdx.x;
  const int tid = threadIdx.x;

  // ---- async stage of the 16x784 activation tile (contiguous 50176 B) ----
  {
    const void* base = (const void*)(agg + (size_t)bt * 16 * 784);
    for (int idx = tid; idx < 3136; idx += 256)
      async_copy_b128(lds_addr(&sA[idx * 4]), (unsigned)(idx * 16), base);
    wait_async0();
  }
  __syncthreads();

  const int wave = tid >> 5;             // = column tile nt (0..7)
  const int lane = tid & 31;
  const int mrow = lane & 15;
  const int ksel = (lane >> 4) * 2;

  const float* aRow = &sA[mrow * 784];
  const float* wCol = w1 + wave * 16 + mrow;

  v8f c0 = {}, c1 = {}, c2 = {}, c3 = {};
  for (int k0 = 0; k0 < 784; k0 += 16) {
    v2f a, bf;
    a.x  = aRow[k0 + ksel];      a.y  = aRow[k0 + ksel + 1];
    bf.x = wCol[(size_t)(k0 + ksel) * 128];
    bf.y = wCol[(size_t)(k0 + ksel + 1) * 128];
    c0 = __builtin_amdgcn_wmma_f32_16x16x4_f32(false, a, false, bf, (short)0, c0, false, false);

    a.x  = aRow[k0 + 4 + ksel];  a.y  = aRow[k0 + 4 + ksel + 1];
    bf.x = wCol[(size_t)(k0 + 4 + ksel) * 128];
    bf.y = wCol[(size_t)(k0 + 4 + ksel + 1) * 128];
    c1 = __builtin_amdgcn_wmma_f32_16x16x4_f32(false, a, false, bf, (short)0, c1, false, false);

    a.x  = aRow[k0 + 8 + ksel];  a.y  = aRow[k0 + 8 + ksel + 1];
    bf.x = wCol[(size_t)(k0 + 8 + ksel) * 128];
    bf.y = wCol[(size_t)(k0 + 8 + ksel + 1) * 128];
    c2 = __builtin_amdgcn_wmma_f32_16x16x4_f32(false, a, false, bf, (short)0, c2, false, false);

    a.x  = aRow[k0 + 12 + ksel]; a.y  = aRow[k0 + 12 + ksel + 1];
    bf.x = wCol[(size_t)(k0 + 12 + ksel) * 128];
    bf.y = wCol[(size_t)(k0 + 12 + ksel + 1) * 128];
    c3 = __builtin_amdgcn_wmma_f32_16x16x4_f32(false, a, false, bf, (short)0, c3, false, false);
  }

  const float bias    = b1[wave * 16 + mrow];
  const int   rowBase = (lane >> 4) * 8;
#pragma unroll
  for (int r = 0; r < 8; ++r) {
    float v = ((c0[r] + c1[r]) + (c2[r] + c3[r])) + bias;
    v = fmaxf(v, 0.f);
    sH[(rowBase + r) * 128 + wave * 16 + mrow] = v;
  }
  __syncthreads();

  // ---- layer 2: 16 rows x 10 logits, K=128 (threads 0..159) ----
  if (tid < 160) {
    const int row = tid / 10, col = tid % 10;
    float o = b2[col];
    const float* hr = &sH[row * 128];
    for (int k = 0; k < 128; ++k) o += hr[k] * w2[k * 10 + col];
    sO[row * 12 + col] = o;
  }
  __syncthreads();

  // ---- log_softmax per row ----
  if (tid < 160) {
    const int row = tid / 10, col = tid % 10;
    const float* lr = &sO[row * 12];
    float m = lr[0];
#pragma unroll
    for (int i = 1; i < 10; ++i) m = fmaxf(m, lr[i]);
    float s = 0.f;
#pragma unroll
    for (int i = 0; i < 10; ++i) s += expf(lr[i] - m);
    out[(size_t)(bt * 16 + row) * 10 + col] = lr[col] - m - logf(s);
  }
}

// ---------------------------------------------------------------------------
extern "C" void kernel_launch(void* const* d_in, const int* in_sizes, int n_in,
                              void* d_out, int out_size, void* d_ws, size_t ws_size,
                              hipStream_t stream) {
  const float* x  = (const float*)d_in[0];
  const float* cw = (const float*)d_in[1];
  const float* cb = (const float*)d_in[2];
  const float* w1 = (const float*)d_in[3];
  const float* b1 = (const float*)d_in[4];
  const float* w2 = (const float*)d_in[5];
  const float* b2 = (const float*)d_in[6];
  float* out = (float*)d_out;

  const int B = in_sizes[0] / 784;       // 2048

  float* agg = (float*)d_ws;             // [B,784]

  k_quanv_agg<<<B, 128, 0, stream>>>(x, cw, cb, agg);
  k_mlp_fused<<<B / 16, 256, 0, stream>>>(agg, w1, b1, w2, b2, out);
}